// MambaBlock_8254927143395
// MI455X (gfx1250) — compile-verified
//
#include <hip/hip_runtime.h>
#include <hip/hip_bf16.h>
#include <math.h>

typedef __attribute__((ext_vector_type(2))) float v2f;
typedef __attribute__((ext_vector_type(4))) float v4f;
typedef __attribute__((ext_vector_type(8))) float v8f;

#define DM   1024
#define DS   16
#define DC   4
#define DI   2048
#define DR   64
#define BATCH 2
#define SEQ  2048
#define ROWS (BATCH * SEQ)   // 4096

// ---------------------------------------------------------------------------
// Generic f32 WMMA GEMM:  C[M,N] = A[M,K] * W[N,K]^T   (weights row-major N x K)
// Block tile 128x64, BK=16, 256 threads = 8 waves.
// Wave w owns rows [w*16, w*16+16) x all 64 cols = four 16x16 accumulators.
// Both As and Bs are stored K-contiguous so every WMMA operand fragment is a
// single aligned ds_load_b64. Row pad +4 (20-float stride) -> lane start banks
// form a perfect stride-4 pattern over all 64 LDS banks (conflict-free b64).
// ---------------------------------------------------------------------------
#define BM 128
#define BN 64
#define BK 16
#define LDP (BK + 4)   // padded row stride

__global__ __launch_bounds__(256) void gemm_f32_wmma(
    const float* __restrict__ A, int lda,
    const float* __restrict__ W, int ldw,
    float* __restrict__ C, int ldc,
    int M, int N, int K)
{
    __shared__ float As[BM][LDP];
    __shared__ float Bs[BN][LDP];

    const int tid  = threadIdx.x;
    const int lane = tid & 31;
    const int wave = tid >> 5;       // 0..7 -> row strip
    const int m16  = lane & 15;      // row (A) / col (B,C) within 16
    const int hi   = lane >> 4;      // half-wave: K pair select / C row split

    const int blockN = blockIdx.x * BN;
    const int blockM = blockIdx.y * BM;

    v8f acc[4] = {};

    for (int k0 = 0; k0 < K; k0 += BK) {
        // --- stage A tile (128 x 16): two float4 per thread, K-contiguous rows
        {
            int r = tid >> 1;              // 0..127
            int c = (tid & 1) * 8;         // 0 or 8
            const float* src = A + (size_t)(blockM + r) * lda + k0 + c;
            v4f v0 = *(const v4f*)(src);
            v4f v1 = *(const v4f*)(src + 4);
            As[r][c + 0] = v0.x; As[r][c + 1] = v0.y; As[r][c + 2] = v0.z; As[r][c + 3] = v0.w;
            As[r][c + 4] = v1.x; As[r][c + 5] = v1.y; As[r][c + 6] = v1.z; As[r][c + 7] = v1.w;
        }
        // --- stage B tile (64 x 16): Bs[n][k] = W[blockN+n][k0+k] (K-contiguous)
        {
            int n  = tid >> 2;             // 0..63
            int c  = (tid & 3) * 4;        // 0,4,8,12
            int gn = blockN + n;
            v4f v = {};
            if (gn < N) v = *(const v4f*)(W + (size_t)gn * ldw + k0 + c);
            Bs[n][c + 0] = v.x; Bs[n][c + 1] = v.y; Bs[n][c + 2] = v.z; Bs[n][c + 3] = v.w;
        }
        __syncthreads();

        // --- 4 K-steps of 4: one A fragment reused across 4 column tiles
        #pragma unroll
        for (int kk = 0; kk < BK; kk += 4) {
            int k2 = kk + 2 * hi;          // lanes 0-15: K=kk,kk+1 ; 16-31: kk+2,kk+3
            v2f a = *(const v2f*)&As[wave * 16 + m16][k2];
            #pragma unroll
            for (int j = 0; j < 4; ++j) {
                v2f b = *(const v2f*)&Bs[j * 16 + m16][k2];
                acc[j] = __builtin_amdgcn_wmma_f32_16x16x4_f32(
                    false, a, false, b, (short)0, acc[j], false, false);
            }
        }
        __syncthreads();
    }

    // --- store: C VGPR v holds row (v + 8*hi), col = lane%16
    #pragma unroll
    for (int j = 0; j < 4; ++j) {
        int col = blockN + j * 16 + m16;
        if (col < N) {
            #pragma unroll
            for (int v = 0; v < 8; ++v) {
                int row = blockM + wave * 16 + v + 8 * hi;
                C[(size_t)row * ldc + col] = acc[j][v];
            }
        }
    }
}

// ---------------------------------------------------------------------------
// Causal depthwise conv (DC=4) + SiLU:  xc = silu(conv(xs) + conv_b)
// xs = xz[..., :DI] (row stride 2*DI)
// ---------------------------------------------------------------------------
__global__ void conv_silu_kernel(const float* __restrict__ xz,
                                 const float* __restrict__ conv_w,
                                 const float* __restrict__ conv_b,
                                 float* __restrict__ xc)
{
    int t = blockIdx.x * blockDim.x + threadIdx.x;
    if (t >= ROWS * DI) return;
    int e   = t % DI;
    int row = t / DI;          // b*SEQ + l
    int l   = row % SEQ;
    float acc = conv_b[e];
    #pragma unroll
    for (int k = 0; k < DC; ++k) {
        int ll = l - (DC - 1) + k;
        if (ll >= 0)
            acc += conv_w[e * DC + k] * xz[(size_t)(row - (DC - 1) + k) * (2 * DI) + e];
    }
    xc[(size_t)row * DI + e] = acc / (1.f + expf(-acc));   // silu
}

// ---------------------------------------------------------------------------
// Selective-scan: 16 lanes per (b,e) hold the 16-wide state h[n].
// Fuses: softplus(delta), dA=exp(delta*A), h = dA*h + delta*B*xc,
//        ys = sum_n h*C (shfl_xor butterfly), y = (ys + xc*D) * silu(z).
// y written in-place over xc (read-before-write within the same wave).
// ---------------------------------------------------------------------------
__global__ void scan_kernel(const float* __restrict__ xz,
                            const float* __restrict__ dbc,
                            const float* __restrict__ dpre,
                            const float* __restrict__ b_dt,
                            const float* __restrict__ A_log,
                            const float* __restrict__ D_param,
                            float* __restrict__ y /* aliases xc */)
{
    int t = blockIdx.x * blockDim.x + threadIdx.x;   // BATCH*DI*16 = 65536
    int n = t & 15;
    int e = (t >> 4) & (DI - 1);
    int b = t >> 15;                                 // t / (DI*16)

    float Aen = -expf(A_log[e * DS + n]);
    float bdt = b_dt[e];
    float Dp  = D_param[e];
    float h   = 0.f;

    for (int l = 0; l < SEQ; ++l) {
        size_t row = (size_t)b * SEQ + l;
        float dp    = dpre[row * DI + e] + bdt;
        float delta = (dp > 20.f) ? dp : log1pf(expf(dp));   // softplus
        float xcv   = y[row * DI + e];                        // xc (pre-overwrite)
        float Bn    = dbc[row * 96 + DR + n];
        float Cn    = dbc[row * 96 + DR + DS + n];
        h = expf(delta * Aen) * h + delta * Bn * xcv;
        float r = h * Cn;
        r += __shfl_xor(r, 1);
        r += __shfl_xor(r, 2);
        r += __shfl_xor(r, 4);
        r += __shfl_xor(r, 8);
        if (n == 0) {
            float z  = xz[row * (size_t)(2 * DI) + DI + e];
            float sz = z / (1.f + expf(-z));
            y[row * DI + e] = (r + xcv * Dp) * sz;
        }
    }
}

// ---------------------------------------------------------------------------
// Residual add + LayerNorm over DM=1024, one block per (b,l) row.
// ---------------------------------------------------------------------------
__global__ __launch_bounds__(256) void resid_ln_kernel(
    const float* __restrict__ hbuf, const float* __restrict__ x,
    const float* __restrict__ g, const float* __restrict__ bta,
    float* __restrict__ out)
{
    __shared__ float red[256];
    int row = blockIdx.x;
    int tid = threadIdx.x;
    float v[4];
    float s = 0.f;
    #pragma unroll
    for (int i = 0; i < 4; ++i) {
        int idx = tid + i * 256;
        v[i] = hbuf[(size_t)row * DM + idx] + x[(size_t)row * DM + idx];
        s += v[i];
    }
    red[tid] = s; __syncthreads();
    for (int o = 128; o > 0; o >>= 1) { if (tid < o) red[tid] += red[tid + o]; __syncthreads(); }
    float mu = red[0] / DM;
    __syncthreads();
    float sq = 0.f;
    #pragma unroll
    for (int i = 0; i < 4; ++i) { float d = v[i] - mu; sq += d * d; }
    red[tid] = sq; __syncthreads();
    for (int o = 128; o > 0; o >>= 1) { if (tid < o) red[tid] += red[tid + o]; __syncthreads(); }
    float inv = rsqrtf(red[0] / DM + 1e-5f);
    #pragma unroll
    for (int i = 0; i < 4; ++i) {
        int idx = tid + i * 256;
        out[(size_t)row * DM + idx] = g[idx] * (v[i] - mu) * inv + bta[idx];
    }
}

// ---------------------------------------------------------------------------
extern "C" void kernel_launch(void* const* d_in, const int* in_sizes, int n_in,
                              void* d_out, int out_size, void* d_ws, size_t ws_size,
                              hipStream_t stream)
{
    (void)in_sizes; (void)n_in; (void)out_size; (void)ws_size;

    const float* x      = (const float*)d_in[0];
    const float* W_in   = (const float*)d_in[1];
    const float* conv_w = (const float*)d_in[2];
    const float* conv_b = (const float*)d_in[3];
    const float* W_x    = (const float*)d_in[4];
    const float* W_dt   = (const float*)d_in[5];
    const float* b_dt   = (const float*)d_in[6];
    const float* A_log  = (const float*)d_in[7];
    const float* D_par  = (const float*)d_in[8];
    const float* W_out  = (const float*)d_in[9];
    const float* ln_g   = (const float*)d_in[10];
    const float* ln_b   = (const float*)d_in[11];
    float* out = (float*)d_out;

    // workspace layout (floats): xz | xc(=y) | dbc | dpre | hbuf  (~153 MB)
    float* ws   = (float*)d_ws;
    float* xz   = ws;                                // ROWS * 2*DI
    float* xc   = xz   + (size_t)ROWS * 2 * DI;      // ROWS * DI   (y in-place)
    float* dbc  = xc   + (size_t)ROWS * DI;          // ROWS * 96
    float* dpre = dbc  + (size_t)ROWS * 96;          // ROWS * DI
    float* hbuf = dpre + (size_t)ROWS * DI;          // ROWS * DM

    // 1. xz = x @ W_in^T      (4096 x 4096 x 1024)
    gemm_f32_wmma<<<dim3((2 * DI) / BN, ROWS / BM), 256, 0, stream>>>(
        x, DM, W_in, DM, xz, 2 * DI, ROWS, 2 * DI, DM);

    // 2. xc = silu(conv(xs) + b)
    conv_silu_kernel<<<(ROWS * DI) / 256, 256, 0, stream>>>(xz, conv_w, conv_b, xc);

    // 3. dbc = xc @ W_x^T     (4096 x 96 x 2048)
    gemm_f32_wmma<<<dim3((96 + BN - 1) / BN, ROWS / BM), 256, 0, stream>>>(
        xc, DI, W_x, DI, dbc, 96, ROWS, 96, DI);

    // 4. dpre = dt @ W_dt^T   (4096 x 2048 x 64), dt = dbc[:, :64] with lda=96
    gemm_f32_wmma<<<dim3(DI / BN, ROWS / BM), 256, 0, stream>>>(
        dbc, 96, W_dt, DR, dpre, DI, ROWS, DI, DR);

    // 5. selective scan -> y (written over xc)
    scan_kernel<<<(BATCH * DI * 16) / 256, 256, 0, stream>>>(
        xz, dbc, dpre, b_dt, A_log, D_par, xc);

    // 6. hbuf = y @ W_out^T   (4096 x 1024 x 2048)
    gemm_f32_wmma<<<dim3(DM / BN, ROWS / BM), 256, 0, stream>>>(
        xc, DI, W_out, DI, hbuf, DM, ROWS, DM, DI);

    // 7. out = LN(hbuf + x)
    resid_ln_kernel<<<ROWS, 256, 0, stream>>>(hbuf, x, ln_g, ln_b, out);
}